// GCNEncoder_12077448036459
// MI455X (gfx1250) — compile-verified
//
#include <hip/hip_runtime.h>
#include <hip/hip_bf16.h>

typedef __attribute__((ext_vector_type(2))) float v2f;
typedef __attribute__((ext_vector_type(4))) float v4f;
typedef __attribute__((ext_vector_type(8))) float v8f;

#define KDIM 128   // inner dim of both GEMMs
#define FW   128   // feature width of intermediate buffers

// ---------------------------------------------------------------------------
// Degree / normalization: dinv[i] = rsqrt(1 + #in-edges(i))
// ---------------------------------------------------------------------------
__global__ void deg_init(float* __restrict__ dinv, int n) {
    int i = blockIdx.x * blockDim.x + threadIdx.x;
    if (i < n) dinv[i] = 1.0f;                 // self-loop contributes 1
}

__global__ void deg_accum(const int* __restrict__ dst, float* __restrict__ dinv, int e) {
    int i = blockIdx.x * blockDim.x + threadIdx.x;
    if (i < e) atomicAdd(&dinv[dst[i]], 1.0f);
}

__global__ void deg_finish(float* __restrict__ dinv, int n) {
    int i = blockIdx.x * blockDim.x + threadIdx.x;
    if (i < n) dinv[i] = rsqrtf(dinv[i]);
}

// ---------------------------------------------------------------------------
// C[nrows x 128] = A[nrows x 128] * B   using V_WMMA_F32_16X16X4_F32.
// blockDim = 256 (8 waves). Block handles a 16-row strip; wave w -> cols [16w,16w+16).
// DUAL=false: all 128 cols from B0 (ld=128).
// DUAL=true : cols 0..63 from B0 (ld=64), cols 64..127 from B1 (ld=64).
//
// Fragments (ISA 7.12.2):
//  A (16x4 f32, 2 VGPR): M=lane&15; lanes 0-15 hold K=k,k+1; lanes 16-31 K=k+2,k+3.
//  B (4x16,    2 VGPR): N=lane&15; lanes 0-15 hold rows k,k+1; lanes 16-31 rows k+2,k+3.
//  C/D (16x16 f32, 8 VGPR): element r -> row r + 8*(lane>>4), col lane&15.
//
// BLD is compile-time so every B access is base + constant-immediate offset
// (no per-iteration 64-bit address math in the WMMA loop).
// ---------------------------------------------------------------------------
template <bool DUAL>
__global__ __launch_bounds__(256)
void gemm_wmma_f32(const float* __restrict__ A,
                   const float* __restrict__ B0,
                   const float* __restrict__ B1,
                   float* __restrict__ C, int nrows)
{
    constexpr int BLD = DUAL ? 64 : 128;
    __shared__ float As[16 * KDIM];            // 8 KB A strip, shared by 8 waves
    const int row0 = blockIdx.x * 16;
    const int tid  = threadIdx.x;

    // Cooperative load of the 16x128 A strip (512 float4, 2 per thread).
    {
        v4f* dstv = (v4f*)As;
        #pragma unroll
        for (int idx = tid; idx < 512; idx += 256) {
            int r  = idx >> 5;                 // strip row (32 float4 per row)
            int c4 = idx & 31;
            int gr = row0 + r; if (gr >= nrows) gr = nrows - 1;   // clamp (no OOB)
            dstv[idx] = ((const v4f*)(A + (size_t)gr * KDIM))[c4];
        }
    }
    __syncthreads();

    const int wave = tid >> 5;                 // 0..7
    const int lane = tid & 31;
    const int col0 = wave * 16;
    const int ln   = lane & 15;
    const int hi   = lane >> 4;                // 0 or 1

    // Hoisted per-lane base pointers; loop indices become constant offsets.
    const float* bp;                           // -> B[2*hi][bcol]
    if (DUAL) {
        const float* Bsel = (col0 < 64) ? B0 : B1;
        const int    bcol = (col0 < 64) ? (col0 + ln) : (col0 - 64 + ln);
        bp = Bsel + (size_t)(2 * hi) * BLD + bcol;
    } else {
        bp = B0 + (size_t)(2 * hi) * BLD + col0 + ln;
    }
    const float* ap = &As[ln * KDIM + 2 * hi]; // -> A[ln][2*hi], 8B aligned

    v8f acc = {};                              // C = 0
    #pragma unroll
    for (int k = 0; k < KDIM; k += 4) {
        v2f a = *(const v2f*)(ap + k);         // A[ln][k+2hi .. k+2hi+1]
        v2f b;
        b.x = bp[(size_t)k * BLD];             // B[k+2hi  ][bcol]  (const imm offset)
        b.y = bp[(size_t)(k + 1) * BLD];       // B[k+2hi+1][bcol]  (const imm offset)
        // (neg_a, A, neg_b, B, c_mod, C, reuse_a, reuse_b)
        acc = __builtin_amdgcn_wmma_f32_16x16x4_f32(
            false, a, false, b, (short)0, acc, false, false);
    }

    // Store 16x16 tile: element r -> row row0 + r + 8*hi, col col0 + ln.
    float* cp = C + (size_t)(row0 + 8 * hi) * FW + col0 + ln;
    #pragma unroll
    for (int r = 0; r < 8; ++r) {
        if (row0 + r + 8 * hi < nrows) cp[(size_t)r * FW] = acc[r];
    }
}

// ---------------------------------------------------------------------------
// out[i,:] = dinv[i]^2 * in[i,:]   (self-loop term; also initializes out)
// ---------------------------------------------------------------------------
__global__ void selfloop_init(const float* __restrict__ in, const float* __restrict__ dinv,
                              float* __restrict__ out, int n)
{
    int t = blockIdx.x * blockDim.x + threadIdx.x;     // over n*128
    if (t < n * FW) {
        int i = t >> 7;
        float d = dinv[i];
        out[t] = d * d * in[t];
    }
}

// ---------------------------------------------------------------------------
// One wave per edge: out[dst,:] += dinv[src]*dinv[dst] * in[src,:]
// lane l handles floats [4l, 4l+4) of the 128-wide row (coalesced 512B gather,
// coalesced f32 atomics that land in L2 — both matrices are L2-resident).
// ---------------------------------------------------------------------------
__global__ __launch_bounds__(256)
void edge_scatter(const int* __restrict__ srcIdx, const int* __restrict__ dstIdx,
                  const float* __restrict__ dinv,
                  const float* __restrict__ in, float* __restrict__ out, int e)
{
    int t = blockIdx.x * blockDim.x + threadIdx.x;
    int edge = t >> 5;
    int lane = t & 31;
    if (edge >= e) return;
    int s = srcIdx[edge];
    int d = dstIdx[edge];
    float nm = dinv[s] * dinv[d];
    v4f v = ((const v4f*)(in + (size_t)s * FW))[lane];
    float* op = out + (size_t)d * FW + lane * 4;
    atomicAdd(op + 0, nm * v.x);
    atomicAdd(op + 1, nm * v.y);
    atomicAdd(op + 2, nm * v.z);
    atomicAdd(op + 3, nm * v.w);
}

// ---------------------------------------------------------------------------
// h = relu(agg + b1)
// ---------------------------------------------------------------------------
__global__ void bias_relu(const float* __restrict__ agg, const float* __restrict__ b,
                          float* __restrict__ out, int n)
{
    int t = blockIdx.x * blockDim.x + threadIdx.x;
    if (t < n * FW) {
        int f = t & (FW - 1);
        out[t] = fmaxf(agg[t] + b[f], 0.0f);
    }
}

// ---------------------------------------------------------------------------
// mu = agg[:, 0:64] + bmu ; logvar = agg[:, 64:128] + blv
// d_out = [mu (N*64) | logvar (N*64)]
// ---------------------------------------------------------------------------
__global__ void final_out(const float* __restrict__ agg,
                          const float* __restrict__ bmu, const float* __restrict__ blv,
                          float* __restrict__ outp, int n)
{
    int t = blockIdx.x * blockDim.x + threadIdx.x;     // over n*64
    if (t < n * 64) {
        int i = t >> 6, f = t & 63;
        const float* row = agg + (size_t)i * FW;
        outp[t]                   = row[f]      + bmu[f];
        outp[(size_t)n * 64 + t]  = row[64 + f] + blv[f];
    }
}

// ---------------------------------------------------------------------------
extern "C" void kernel_launch(void* const* d_in, const int* in_sizes, int n_in,
                              void* d_out, int out_size, void* d_ws, size_t ws_size,
                              hipStream_t stream)
{
    const float* x    = (const float*)d_in[0];
    const int*   ei   = (const int*)  d_in[1];
    const float* W1   = (const float*)d_in[2];
    const float* b1   = (const float*)d_in[3];
    const float* Wmu  = (const float*)d_in[4];
    const float* bmu  = (const float*)d_in[5];
    const float* Wlv  = (const float*)d_in[6];
    const float* blv  = (const float*)d_in[7];

    const int N = in_sizes[0] / KDIM;          // 100,000
    const int E = in_sizes[1] / 2;             // 1,600,000
    const int* srcIdx = ei;
    const int* dstIdx = ei + E;

    // Workspace: dinv[N] | bufA[N*128] | bufB[N*128]  (~103 MB)
    size_t dinv_bytes = (((size_t)N * 4) + 255) & ~(size_t)255;
    float* dinv = (float*)d_ws;
    float* bufA = (float*)((char*)d_ws + dinv_bytes);
    float* bufB = bufA + (size_t)N * FW;

    const int TB = 256;
    dim3 blk(TB);
    int gN   = (N + TB - 1) / TB;
    int gE   = (E + TB - 1) / TB;
    int gNF  = (N * FW + TB - 1) / TB;
    int gNZ  = (N * 64 + TB - 1) / TB;
    int gEw  = (int)(((size_t)E * 32 + TB - 1) / TB);  // wave per edge
    int gGemm = (N + 15) / 16;

    // 1) normalization
    deg_init  <<<gN, blk, 0, stream>>>(dinv, N);
    deg_accum <<<gE, blk, 0, stream>>>(dstIdx, dinv, E);
    deg_finish<<<gN, blk, 0, stream>>>(dinv, N);

    // 2) layer 1: bufA = X @ W1 (WMMA fp32)
    gemm_wmma_f32<false><<<gGemm, blk, 0, stream>>>(x, W1, nullptr, bufA, N);

    // 3) aggregate: bufB = D^-1/2 (A+I) D^-1/2 bufA
    selfloop_init<<<gNF, blk, 0, stream>>>(bufA, dinv, bufB, N);
    edge_scatter <<<gEw, blk, 0, stream>>>(srcIdx, dstIdx, dinv, bufA, bufB, E);

    // 4) bufA = relu(bufB + b1)
    bias_relu<<<gNF, blk, 0, stream>>>(bufB, b1, bufA, N);

    // 5) layer 2 (fused mu|logvar): bufB = bufA @ [Wmu | Wlv]
    gemm_wmma_f32<true><<<gGemm, blk, 0, stream>>>(bufA, Wmu, Wlv, bufB, N);

    // 6) aggregate into bufA
    selfloop_init<<<gNF, blk, 0, stream>>>(bufB, dinv, bufA, N);
    edge_scatter <<<gEw, blk, 0, stream>>>(srcIdx, dstIdx, dinv, bufB, bufA, E);

    // 7) biases + split halves into d_out
    final_out<<<gNZ, blk, 0, stream>>>(bufA, bmu, blv, (float*)d_out, N);
}